// OrderInvariantSeq2SeqCrossEntropy_26620207300999
// MI455X (gfx1250) — compile-verified
//
#include <hip/hip_runtime.h>
#include <hip/hip_bf16.h>
#include <math.h>

#define PAD_IDX     0
#define NB_BLOCKS   5
#define BLOCK_SIZE  5
#define SEQ         25          // NB_BLOCKS * BLOCK_SIZE
#define VOCAB       32000
#define BATCH       256
#define SCALE_F     0.05f

#define NTHREADS    256
#define NVEC        (VOCAB / 4)                       // 8000 float4 per row
#define NCHUNK      ((NVEC + NTHREADS - 1) / NTHREADS) // 32 chunks
#define STAGE_BUFS  4                                  // 4-deep LDS ring

// ---- CDNA5 async global->LDS path (probe with __has_builtin) --------------
#if defined(__HIP_DEVICE_COMPILE__) && __has_builtin(__builtin_amdgcn_global_load_async_to_lds_b128)
#define HAVE_ASYNC_LDS 1
#else
#define HAVE_ASYNC_LDS 0
#endif

#if defined(__HIP_DEVICE_COMPILE__) && __has_builtin(__builtin_amdgcn_s_wait_asynccnt)
#define WAIT_ASYNC(n) __builtin_amdgcn_s_wait_asynccnt(n)
#else
#define WAIT_ASYNC(n) asm volatile("s_wait_asynccnt %0" :: "i"(n) : "memory")
#endif

typedef int v4i __attribute__((ext_vector_type(4)));
typedef __attribute__((address_space(1))) v4i* gv4i_p;   // global (AS1) int4*
typedef __attribute__((address_space(3))) v4i* lv4i_p;   // LDS (AS3) int4*

// ---------------------------------------------------------------------------
// Kernel 1: per-row online log-sum-exp + gathered logits.
//   grid = (SEQ, BATCH), block = 256 threads (8 wave32).
//   Outputs: ws_nll[row*5 + t]  (masked -logp at the 5 candidate targets)
//            ws_push[row]       (-logp at PAD_IDX)
// ---------------------------------------------------------------------------
__global__ __launch_bounds__(NTHREADS)
void oice_rowpass(const float* __restrict__ yhat,
                  const int*   __restrict__ target,
                  float* __restrict__ ws_nll,
                  float* __restrict__ ws_push) {
  const int sq  = blockIdx.x;          // 0..24
  const int b   = blockIdx.y;          // 0..255
  const int tid = threadIdx.x;

  const float*  row  = yhat + ((size_t)b * (SEQ + 1) + sq) * VOCAB;
  const float4* rowv = reinterpret_cast<const float4*>(row);

  float m = -3.402823466e+38f;  // running max
  float s = 0.0f;               // running sum of exp(x - m)

  // One exp per element: E = exp(min(m,x) - max(m,x)).
  auto upd = [&](float x) {
    float nm = fmaxf(m, x);
    float E  = __expf(fminf(m, x) - nm);
    s = (x > m) ? __fmaf_rn(s, E, 1.0f) : (s + E);
    m = nm;
  };

#if HAVE_ASYNC_LDS
  // Stream the row through LDS with the CDNA5 async copy engine.
  // Each wave issues/consumes only its own 32x16B slice of each chunk,
  // so ASYNCcnt bookkeeping is purely per-wave (no barriers in the loop).
  __shared__ float4 stage[STAGE_BUFS][NTHREADS];
  auto issue = [&](int k) {
    int v = k * NTHREADS + tid;
    v = (v < NVEC - 1) ? v : (NVEC - 1);   // clamp tail: all lanes always issue
    __builtin_amdgcn_global_load_async_to_lds_b128(
        (gv4i_p)(rowv + v),
        (lv4i_p)&stage[k & (STAGE_BUFS - 1)][tid],
        0, 0);
  };
  issue(0); issue(1); issue(2);
  for (int k = 0; k < NCHUNK; ++k) {
    if      (k + 3 < NCHUNK) { issue(k + 3); WAIT_ASYNC(3); }
    else if (k + 2 < NCHUNK) { WAIT_ASYNC(2); }
    else if (k + 1 < NCHUNK) { WAIT_ASYNC(1); }
    else                     { WAIT_ASYNC(0); }
    asm volatile("" ::: "memory");
    int v = k * NTHREADS + tid;
    if (v < NVEC) {
      float4 x = stage[k & (STAGE_BUFS - 1)][tid];   // ds_load_b128
      upd(x.x); upd(x.y); upd(x.z); upd(x.w);
    }
  }
#else
  // Fallback: direct b128 global loads, online softmax.
  for (int v = tid; v < NVEC; v += NTHREADS) {
    float4 x = rowv[v];
    upd(x.x); upd(x.y); upd(x.z); upd(x.w);
  }
#endif

  // Wave32 reduction of (m, s).
  for (int off = 16; off > 0; off >>= 1) {
    float mo = __shfl_xor(m, off, 32);
    float so = __shfl_xor(s, off, 32);
    float nm = fmaxf(m, mo);
    s = s * __expf(m - nm) + so * __expf(mo - nm);
    m = nm;
  }

  __shared__ float red_m[NTHREADS / 32];
  __shared__ float red_s[NTHREADS / 32];
  __shared__ float s_lse;
  const int wave = tid >> 5, lane = tid & 31;
  if (lane == 0) { red_m[wave] = m; red_s[wave] = s; }
  __syncthreads();
  if (tid == 0) {
    float M = red_m[0], S = red_s[0];
    #pragma unroll
    for (int w = 1; w < NTHREADS / 32; ++w) {
      float nm = fmaxf(M, red_m[w]);
      S = S * __expf(M - nm) + red_s[w] * __expf(red_m[w] - nm);
      M = nm;
    }
    s_lse = M + __logf(S);
  }
  __syncthreads();
  const float lse = s_lse;

  const size_t rowid = (size_t)b * SEQ + sq;
  if (tid < NB_BLOCKS) {
    // candidate target for block t at this seq position
    int idx = target[(size_t)b * (SEQ + 2) + 1 + tid * BLOCK_SIZE + (sq % BLOCK_SIZE)];
    float v = 0.0f;
    if (idx != PAD_IDX) v = -(row[idx] - lse);
    ws_nll[rowid * NB_BLOCKS + tid] = v;
  } else if (tid == NB_BLOCKS) {
    ws_push[rowid] = -(row[PAD_IDX] - lse);
  }
}

// ---------------------------------------------------------------------------
// Kernel 2: combine into the final scalar. One 256-thread block (tid == b).
// ---------------------------------------------------------------------------
__global__ __launch_bounds__(BATCH)
void oice_finalize(const float* __restrict__ ws_nll,
                   const float* __restrict__ ws_push,
                   const int*   __restrict__ target,
                   float* __restrict__ out) {
  __shared__ float blr[NB_BLOCKS][BATCH];
  __shared__ float pr[BATCH];
  __shared__ int   act[NB_BLOCKS];
  const int tid = threadIdx.x;   // == batch index b
  if (tid < NB_BLOCKS) act[tid] = 0;
  __syncthreads();

  // push-loss partial for this batch row
  float p = 0.0f;
  for (int s = 0; s < SEQ; ++s) p += ws_push[(size_t)tid * SEQ + s];
  pr[tid] = p;

  const int b = tid;
  #pragma unroll
  for (int t = 0; t < NB_BLOCKS; ++t) {
    float minv = 3.402823466e+38f;
    #pragma unroll
    for (int j = 0; j < NB_BLOCKS; ++j) {
      float acc = 0.0f;
      #pragma unroll
      for (int i = 0; i < BLOCK_SIZE; ++i) {
        int srow = b * SEQ + (j * BLOCK_SIZE + i);
        acc += ws_nll[(size_t)srow * NB_BLOCKS + t];
      }
      minv = fminf(minv, acc * (1.0f / BLOCK_SIZE));
    }
    blr[t][tid] = minv;
    int anynz = 0;
    #pragma unroll
    for (int i = 0; i < BLOCK_SIZE; ++i)
      anynz |= (target[(size_t)b * (SEQ + 2) + 1 + t * BLOCK_SIZE + i] != PAD_IDX) ? 1 : 0;
    if (anynz) atomicOr(&act[t], 1);
  }

  for (int st = BATCH / 2; st > 0; st >>= 1) {
    __syncthreads();
    if (tid < st) {
      pr[tid] += pr[tid + st];
      #pragma unroll
      for (int t = 0; t < NB_BLOCKS; ++t) blr[t][tid] += blr[t][tid + st];
    }
  }
  __syncthreads();

  if (tid == 0) {
    float inc = 1.0f, num = 0.0f, den = 0.0f;
    #pragma unroll
    for (int t = 0; t < NB_BLOCKS; ++t) {
      inc *= act[t] ? 1.0f : 0.0f;
      num += (blr[t][0] * (1.0f / BATCH)) * inc;
      den += inc;
    }
    float order_loss = num / fmaxf(den, 1.0f);
    float push_loss  = pr[0] * (1.0f / (BATCH * SEQ));
    out[0] = order_loss + SCALE_F * push_loss;
  }
}

// ---------------------------------------------------------------------------
extern "C" void kernel_launch(void* const* d_in, const int* in_sizes, int n_in,
                              void* d_out, int out_size, void* d_ws, size_t ws_size,
                              hipStream_t stream) {
  const float* yhat   = (const float*)d_in[0];  // (256, 26, 32000) f32
  const int*   target = (const int*)d_in[1];    // (256, 27) i32

  float* ws      = (float*)d_ws;
  float* ws_nll  = ws;                               // BATCH*SEQ*NB_BLOCKS = 32000 floats
  float* ws_push = ws + (size_t)BATCH * SEQ * NB_BLOCKS;  // + 6400 floats

  dim3 grid(SEQ, BATCH);
  oice_rowpass<<<grid, NTHREADS, 0, stream>>>(yhat, target, ws_nll, ws_push);
  oice_finalize<<<1, BATCH, 0, stream>>>(ws_nll, ws_push, target, (float*)d_out);
}